// HeteroConvModel_29171417874768
// MI455X (gfx1250) — compile-verified
//
#include <hip/hip_runtime.h>
#include <hip/hip_bf16.h>

#define N_MAT  200000
#define N_ELEM 50000
#define F_MAT  64
#define DD     128
#define HH     512
#define EDGES  600000
#define OUT_C  8

// LDS row strides padded to ≡4 (mod 64 banks): conflict-free WMMA A-fragment loads
#define LXP    132   // DD + 4
#define LHP    516   // HH + 4
#define LFP    68    // F_MAT + 4

typedef float v2f __attribute__((ext_vector_type(2)));
typedef float v8f __attribute__((ext_vector_type(8)));

static __device__ __forceinline__ v8f wmma4(v2f a, v2f b, v8f c) {
    // D = A(16x4) * B(4x16) + C(16x16), fp32, wave32
    return __builtin_amdgcn_wmma_f32_16x16x4_f32(
        /*neg_a=*/false, a, /*neg_b=*/false, b,
        /*c_mod=*/(short)0, c, /*reuse_a=*/false, /*reuse_b=*/false);
}

// ---------------------------------------------------------------------------
// Input layer: xm[r] = x_mat[r] @ W_in + b_in + emb_mat[r]
// grid = N_MAT/16, block = 256 (8 waves; wave w -> N-tile n0 = 16*w)
// ---------------------------------------------------------------------------
__global__ __launch_bounds__(256) void input_kernel(
    const float* __restrict__ x_mat, const float* __restrict__ W_in,
    const float* __restrict__ b_in,  const float* __restrict__ emb,
    float* __restrict__ xm)
{
    __shared__ float lx[16 * LFP];
    const int tid = threadIdx.x;
    const long rowbase = (long)blockIdx.x * 16;
    for (int i = 0; i < 4; ++i) {
        int idx = tid + i * 256;
        int row = idx >> 6, col = idx & 63;
        lx[row * LFP + col] = x_mat[(rowbase + row) * F_MAT + col];
    }
    __syncthreads();

    const int wave = tid >> 5, lane = tid & 31;
    const int ln = lane & 15, kh = lane >> 4;
    const int n0 = wave * 16;
    v8f acc = {};
    for (int k0 = 0; k0 < F_MAT; k0 += 4) {
        v2f a; a.x = lx[ln * LFP + k0 + 2 * kh];
               a.y = lx[ln * LFP + k0 + 2 * kh + 1];
        v2f b; b.x = W_in[(k0 + 2 * kh) * DD + n0 + ln];
               b.y = W_in[(k0 + 2 * kh + 1) * DD + n0 + ln];
        acc = wmma4(a, b, acc);
    }
    const float bias = b_in[n0 + ln];
    for (int r = 0; r < 8; ++r) {
        long row = rowbase + r + 8 * kh;
        xm[row * DD + n0 + ln] = acc[r] + bias + emb[row * DD + n0 + ln];
    }
}

// ---------------------------------------------------------------------------
// Fused pre-LN FFN: out = relu(LN(x) @ W1 + b1) @ W2 + b2 (in-place safe)
// 32-row M-tile per block (B fragments reused across 2 M-subtiles).
// grid = ceil(nrows/32), block = 256. Tail rows clamped/guarded.
// ---------------------------------------------------------------------------
__global__ __launch_bounds__(256) void ffn_kernel(
    const float* __restrict__ x_in, float* __restrict__ x_out,
    const float* __restrict__ g,  const float* __restrict__ bb,
    const float* __restrict__ W1, const float* __restrict__ b1,
    const float* __restrict__ W2, const float* __restrict__ b2,
    const long nrows)
{
    __shared__ float lx[32 * LXP];   // ~16.9 KB (LayerNorm'd input tile)
    __shared__ float lh[32 * LHP];   // ~66 KB   (hidden tile)
    const int tid = threadIdx.x;
    const long rowbase = (long)blockIdx.x * 32;

    for (int i = 0; i < 16; ++i) {
        int idx = tid + i * 256;
        int row = idx >> 7, col = idx & 127;
        long grow = rowbase + row;
        if (grow >= nrows) grow = nrows - 1;   // clamp stays inside this block's rows
        lx[row * LXP + col] = x_in[grow * DD + col];
    }
    __syncthreads();

    // LayerNorm: 8 lanes per row (32 rows x 8 lanes = 256 threads)
    {
        const int row = tid >> 3, sub = tid & 7;
        float v[16], s = 0.f, ss = 0.f;
        for (int j = 0; j < 16; ++j) {
            v[j] = lx[row * LXP + sub + j * 8];
            s += v[j]; ss += v[j] * v[j];
        }
        for (int m = 1; m < 8; m <<= 1) {
            s  += __shfl_xor(s,  m, 32);
            ss += __shfl_xor(ss, m, 32);
        }
        const float mean = s * (1.f / DD);
        const float var  = ss * (1.f / DD) - mean * mean;
        const float rinv = rsqrtf(var + 1e-5f);
        for (int j = 0; j < 16; ++j) {
            int c = sub + j * 8;
            lx[row * LXP + c] = (v[j] - mean) * rinv * g[c] + bb[c];
        }
    }
    __syncthreads();

    const int wave = tid >> 5, lane = tid & 31;
    const int ln = lane & 15, kh = lane >> 4;

    // GEMM1: (32xD) @ (DxH) -> relu -> lh ; wave owns 64 cols x 2 M-subtiles
    {
        v8f acc[2][4] = {};
        for (int k0 = 0; k0 < DD; k0 += 4) {
            v2f a0; a0.x = lx[ln * LXP + k0 + 2 * kh];
                    a0.y = lx[ln * LXP + k0 + 2 * kh + 1];
            v2f a1; a1.x = lx[(16 + ln) * LXP + k0 + 2 * kh];
                    a1.y = lx[(16 + ln) * LXP + k0 + 2 * kh + 1];
            for (int t = 0; t < 4; ++t) {
                const int n0 = wave * 64 + t * 16;
                v2f b; b.x = W1[(k0 + 2 * kh) * HH + n0 + ln];
                       b.y = W1[(k0 + 2 * kh + 1) * HH + n0 + ln];
                acc[0][t] = wmma4(a0, b, acc[0][t]);
                acc[1][t] = wmma4(a1, b, acc[1][t]);
            }
        }
        for (int mt = 0; mt < 2; ++mt)
            for (int t = 0; t < 4; ++t) {
                const int n0 = wave * 64 + t * 16;
                const float bias = b1[n0 + ln];
                for (int r = 0; r < 8; ++r) {
                    float val = acc[mt][t][r] + bias;
                    lh[(mt * 16 + r + 8 * kh) * LHP + n0 + ln] = val > 0.f ? val : 0.f;
                }
            }
    }
    __syncthreads();

    // GEMM2: (32xH) @ (HxD) + b2 ; wave owns one 16-col tile x 2 M-subtiles
    {
        const int n0 = wave * 16;
        v8f acc0 = {}, acc1 = {};
        for (int k0 = 0; k0 < HH; k0 += 4) {
            v2f b; b.x = W2[(k0 + 2 * kh) * DD + n0 + ln];
                   b.y = W2[(k0 + 2 * kh + 1) * DD + n0 + ln];
            v2f a0; a0.x = lh[ln * LHP + k0 + 2 * kh];
                    a0.y = lh[ln * LHP + k0 + 2 * kh + 1];
            v2f a1; a1.x = lh[(16 + ln) * LHP + k0 + 2 * kh];
                    a1.y = lh[(16 + ln) * LHP + k0 + 2 * kh + 1];
            acc0 = wmma4(a0, b, acc0);
            acc1 = wmma4(a1, b, acc1);
        }
        const float bias = b2[n0 + ln];
        for (int r = 0; r < 8; ++r) {
            long row = rowbase + r + 8 * kh;
            if (row < nrows) x_out[row * DD + n0 + ln] = acc0[r] + bias;
            row = rowbase + 16 + r + 8 * kh;
            if (row < nrows) x_out[row * DD + n0 + ln] = acc1[r] + bias;
        }
    }
}

// ---------------------------------------------------------------------------
// Edge scatter-add (mean aggregation numerator + counts).
// One thread per (edge, feature); both gather table and aggr are L2-resident.
// grid = EDGES*DD/256 = 300000
// ---------------------------------------------------------------------------
__global__ __launch_bounds__(256) void scatter_kernel(
    const float* __restrict__ xsrc, const int* __restrict__ src,
    const int* __restrict__ dst, float* __restrict__ aggr,
    float* __restrict__ cnt)
{
    const long idx = (long)blockIdx.x * 256 + threadIdx.x;
    const int e = (int)(idx >> 7);
    const int f = (int)(idx & 127);
    const int s = src[e], d = dst[e];
    unsafeAtomicAdd(&aggr[(long)d * DD + f], xsrc[(long)s * DD + f]);
    if (f == 0) unsafeAtomicAdd(&cnt[d], 1.0f);
}

// ---------------------------------------------------------------------------
// SAGE linear: aggr[r] <- (aggr[r]/max(cnt,1)) @ Wl + bl + xdst[r] @ Wr
// in-place on aggr; 32-row tile; grid = N_MAT/32, block = 256
// ---------------------------------------------------------------------------
__global__ __launch_bounds__(256) void sage_kernel(
    float* __restrict__ aggr, const float* __restrict__ cnt,
    const float* __restrict__ xdst,
    const float* __restrict__ Wl, const float* __restrict__ bl,
    const float* __restrict__ Wr)
{
    __shared__ float la[32 * LXP];
    __shared__ float lr[32 * LXP];
    const int tid = threadIdx.x;
    const long rowbase = (long)blockIdx.x * 32;
    for (int i = 0; i < 16; ++i) {
        int idx = tid + i * 256;
        int row = idx >> 7, col = idx & 127;
        float c = cnt[rowbase + row];
        c = c > 1.f ? c : 1.f;
        la[row * LXP + col] = aggr[(rowbase + row) * DD + col] / c;
        lr[row * LXP + col] = xdst[(rowbase + row) * DD + col];
    }
    __syncthreads();

    const int wave = tid >> 5, lane = tid & 31;
    const int ln = lane & 15, kh = lane >> 4;
    const int n0 = wave * 16;
    v8f acc0 = {}, acc1 = {};
    for (int k0 = 0; k0 < DD; k0 += 4) {
        v2f b; b.x = Wl[(k0 + 2 * kh) * DD + n0 + ln];
               b.y = Wl[(k0 + 2 * kh + 1) * DD + n0 + ln];
        v2f a0; a0.x = la[ln * LXP + k0 + 2 * kh];
                a0.y = la[ln * LXP + k0 + 2 * kh + 1];
        v2f a1; a1.x = la[(16 + ln) * LXP + k0 + 2 * kh];
                a1.y = la[(16 + ln) * LXP + k0 + 2 * kh + 1];
        acc0 = wmma4(a0, b, acc0);
        acc1 = wmma4(a1, b, acc1);
    }
    for (int k0 = 0; k0 < DD; k0 += 4) {
        v2f b; b.x = Wr[(k0 + 2 * kh) * DD + n0 + ln];
               b.y = Wr[(k0 + 2 * kh + 1) * DD + n0 + ln];
        v2f a0; a0.x = lr[ln * LXP + k0 + 2 * kh];
                a0.y = lr[ln * LXP + k0 + 2 * kh + 1];
        v2f a1; a1.x = lr[(16 + ln) * LXP + k0 + 2 * kh];
                a1.y = lr[(16 + ln) * LXP + k0 + 2 * kh + 1];
        acc0 = wmma4(a0, b, acc0);
        acc1 = wmma4(a1, b, acc1);
    }
    const float bias = bl[n0 + ln];
    for (int r = 0; r < 8; ++r) {
        long row = rowbase + r + 8 * kh;
        aggr[row * DD + n0 + ln] = acc0[r] + bias;
        row = rowbase + 16 + r + 8 * kh;
        aggr[row * DD + n0 + ln] = acc1[r] + bias;
    }
}

// ---------------------------------------------------------------------------
// Output head: out = x @ W_out + b_out   (N=8; LDS-staged VALU, padded tile)
// grid = N_MAT/32, block = 256 (32 rows x 8 outputs)
// ---------------------------------------------------------------------------
__global__ __launch_bounds__(256) void head_kernel(
    const float* __restrict__ x, const float* __restrict__ W,
    const float* __restrict__ b, float* __restrict__ out)
{
    __shared__ float lx[32 * LXP];     // ~16.9 KB
    __shared__ float lw[DD * OUT_C];   // 4 KB
    const int tid = threadIdx.x;
    const long rowbase = (long)blockIdx.x * 32;
    for (int i = 0; i < 16; ++i) {
        int idx = tid + i * 256;
        int row = idx >> 7, col = idx & 127;
        lx[row * LXP + col] = x[(rowbase + row) * DD + col];
    }
    for (int i = tid; i < DD * OUT_C; i += 256)
        lw[i] = W[i];
    __syncthreads();
    const int row = tid >> 3, o = tid & 7;
    float s = b[o];
    for (int k = 0; k < DD; ++k)
        s += lx[row * LXP + k] * lw[k * OUT_C + o];
    out[(rowbase + row) * OUT_C + o] = s;
}

// ---------------------------------------------------------------------------
// Launcher. Input order (setup_inputs dict order, params flattened depth-first):
// 0 x_mat | 1 src_me | 2 dst_me | 3 src_em | 4 dst_em
// 5 emb_mat | 6 emb_elem | 7 W_in | 8 b_in
// 9..14  ffn1_mat  {g,b,W1,b1,W2,b2} | 15..20 ffn1_elem
// 21..23 sage_me {Wl,bl,Wr} | 24..26 sage_em {Wl,bl,Wr}
// 27..32 ffn2_mat | 33..38 ffn2_elem | 39 W_out | 40 b_out
// Only the subgraph reaching the output (material branch via sage_em) is run.
// ---------------------------------------------------------------------------
extern "C" void kernel_launch(void* const* d_in, const int* in_sizes, int n_in,
                              void* d_out, int out_size, void* d_ws, size_t ws_size,
                              hipStream_t stream) {
    const float* x_mat    = (const float*)d_in[0];
    const int*   src_em   = (const int*)d_in[3];
    const int*   dst_em   = (const int*)d_in[4];
    const float* emb_mat  = (const float*)d_in[5];
    const float* emb_elem = (const float*)d_in[6];
    const float* W_in     = (const float*)d_in[7];
    const float* b_in     = (const float*)d_in[8];

    const float* f1m[6]; for (int i = 0; i < 6; ++i) f1m[i] = (const float*)d_in[9 + i];
    const float* f1e[6]; for (int i = 0; i < 6; ++i) f1e[i] = (const float*)d_in[15 + i];
    const float* sWl = (const float*)d_in[24];
    const float* sbl = (const float*)d_in[25];
    const float* sWr = (const float*)d_in[26];
    const float* f2m[6]; for (int i = 0; i < 6; ++i) f2m[i] = (const float*)d_in[27 + i];
    const float* W_out = (const float*)d_in[39];
    const float* b_out = (const float*)d_in[40];

    // workspace layout
    float* xm   = (float*)d_ws;                       // N_MAT*DD
    float* xe   = xm + (size_t)N_MAT * DD;            // N_ELEM*DD
    float* aggr = xe + (size_t)N_ELEM * DD;           // N_MAT*DD
    float* cnt  = aggr + (size_t)N_MAT * DD;          // N_MAT

    // 1. input layer (material nodes)
    input_kernel<<<N_MAT / 16, 256, 0, stream>>>(x_mat, W_in, b_in, emb_mat, xm);
    // 2. ffn1 on both node types (element input = embedding table directly)
    ffn_kernel<<<N_MAT / 32, 256, 0, stream>>>(xm, xm,
        f1m[0], f1m[1], f1m[2], f1m[3], f1m[4], f1m[5], (long)N_MAT);
    ffn_kernel<<<(N_ELEM + 31) / 32, 256, 0, stream>>>(emb_elem, xe,
        f1e[0], f1e[1], f1e[2], f1e[3], f1e[4], f1e[5], (long)N_ELEM);
    // 3. mean aggregation elem -> mat (edge type "em")
    hipMemsetAsync(aggr, 0, (size_t)N_MAT * DD * sizeof(float), stream);
    hipMemsetAsync(cnt,  0, (size_t)N_MAT * sizeof(float), stream);
    scatter_kernel<<<(EDGES * DD) / 256, 256, 0, stream>>>(xe, src_em, dst_em, aggr, cnt);
    // 4. SAGE linear (in-place on aggr)
    sage_kernel<<<N_MAT / 32, 256, 0, stream>>>(aggr, cnt, xm, sWl, sbl, sWr);
    // 5. ffn2 on material nodes (in-place)
    ffn_kernel<<<N_MAT / 32, 256, 0, stream>>>(aggr, aggr,
        f2m[0], f2m[1], f2m[2], f2m[3], f2m[4], f2m[5], (long)N_MAT);
    // 6. output head
    head_kernel<<<N_MAT / 32, 256, 0, stream>>>(aggr, W_out, b_out, (float*)d_out);
}